// ResidualVQ_45148696216861
// MI455X (gfx1250) — compile-verified
//
#include <hip/hip_runtime.h>

// ResidualVQ forward for MI455X (gfx1250, wave32).
// out[n] = codebook[argmin_k (||c_k||^2 - 2 x_n . c_k)]
//
// GEMM tile = codes(M=16) x tokens(N=16), accumulated over D=128 with
// V_WMMA_F32_16X16X4_F32 (32 WMMAs per tile). B (tokens) lives in registers
// for the whole kernel; A (codebook) streams through double-buffered LDS via
// the CDNA5 async global->LDS path (ASYNCcnt), overlapping fill with WMMA.

typedef __attribute__((ext_vector_type(2))) float v2f;
typedef __attribute__((ext_vector_type(4))) float v4f;
typedef __attribute__((ext_vector_type(8))) float v8f;
typedef __attribute__((ext_vector_type(4))) int   v4i;

typedef __attribute__((address_space(1))) v4i* as1_v4i_p;
typedef __attribute__((address_space(3))) v4i* as3_v4i_p;

#define N_TOK 32768
#define DIM   128
#define K_CB  1024
#define KC    64          // codebook rows per LDS stage buffer
#define ROWP  132         // padded LDS row stride in floats (132 % 64 == 4)
#define WAVES 8
#define TPB   256         // 8 wave32 per workgroup
#define TOK_PER_WG (WAVES * 16)

// ---- CDNA5 async global->LDS copy of one 16-byte chunk --------------------
__device__ __forceinline__ void async_cp_b128(const float* g, float* l) {
#if __has_builtin(__builtin_amdgcn_global_load_async_to_lds_b128)
    __builtin_amdgcn_global_load_async_to_lds_b128(
        (as1_v4i_p)(unsigned long long)g,
        (as3_v4i_p)(unsigned int)(unsigned long long)l,
        0, 0);
#else
    // GV mode: vdst = 32-bit LDS byte address, vaddr = 64-bit global address.
    unsigned int lds_off = (unsigned int)(unsigned long long)l;
    asm volatile("global_load_async_to_lds_b128 %0, %1, off"
                 :: "v"(lds_off), "v"(g) : "memory");
#endif
}

__device__ __forceinline__ void wait_async0() {
#if __has_builtin(__builtin_amdgcn_s_wait_asynccnt)
    __builtin_amdgcn_s_wait_asynccnt(0);
#else
    asm volatile("s_wait_asynccnt 0x0" ::: "memory");
#endif
}

// Issue this thread's share of one 64-row codebook stage (8 x b128).
__device__ __forceinline__ void stage_issue(const float* __restrict__ cb,
                                            int kc, float* lbuf, int tid) {
#pragma unroll
    for (int t = 0; t < (KC * DIM / 4) / TPB; ++t) {  // 8 iterations
        int f   = tid + t * TPB;                      // float4 index, 32/row
        int row = f >> 5;
        int c4  = f & 31;
        const float* g = cb + (size_t)(kc + row) * DIM + c4 * 4;
        float*       l = lbuf + row * ROWP + c4 * 4;
        async_cp_b128(g, l);
    }
}

__global__ __launch_bounds__(TPB) void
vq_argmin_gather(const float* __restrict__ x,
                 const float* __restrict__ cb,
                 float* __restrict__ out) {
    __shared__ float lds_cn[K_CB];            // all ||c||^2, persistent (4 KB)
    __shared__ float lds_cb[2][KC * ROWP];    // double buffer (~66 KB)

    const int tid  = threadIdx.x;
    const int lane = tid & 31;
    const int wave = tid >> 5;
    const int lo   = lane & 15;               // token column within wave tile
    const int hi   = lane >> 4;               // lane-half selector
    const int koff = 2 * hi;                  // K-offset inside 4-wide slice

    // Token this lane owns (matrix B column). 16 tokens per wave.
    const int tok = blockIdx.x * TOK_PER_WG + wave * 16 + lo;

    // ---- Kick off stage 0 fill (async, overlaps with prologue work) ------
    stage_issue(cb, 0, &lds_cb[0][0], tid);

    // ---- Load x tile into resident B fragments ---------------------------
    // WMMA f32 16x16x4 B-frag layout (mirrors documented A-frag):
    //   lanes 0-15 hold K = 0,1 (vgpr0,vgpr1); lanes 16-31 hold K = 2,3.
    v2f bfrag[DIM / 4];
    {
        const float* xrow = x + (size_t)tok * DIM;
#pragma unroll
        for (int j = 0; j < DIM / 4; ++j)
            bfrag[j] = *(const v2f*)(xrow + 4 * j + koff);
    }

    // ---- Precompute all ||c||^2 into LDS once (4 rows per thread) --------
    for (int c = tid; c < K_CB; c += TPB) {   // uniform: exactly 4 iters
        const v4f* row = (const v4f*)(cb + (size_t)c * DIM);
        float s = 0.f;
#pragma unroll
        for (int i = 0; i < DIM / 4; ++i) {
            v4f v = row[i];
            s += v.x * v.x + v.y * v.y + v.z * v.z + v.w * v.w;
        }
        lds_cn[c] = s;
    }

    wait_async0();          // stage-0 buffer filled (this wave's share)
    __syncthreads();        // + everyone's share, + lds_cn visible

    float best = 3.4e38f;
    int   bidx = 0;

#pragma unroll 1
    for (int stage = 0; stage < K_CB / KC; ++stage) {
        const int kc = stage * KC;
        const int p  = stage & 1;

        // Stream next stage into the other buffer while we compute.
        if (kc + KC < K_CB)
            stage_issue(cb, kc + KC, &lds_cb[p ^ 1][0], tid);

        const float* buf = &lds_cb[p][0];

#pragma unroll
        for (int sub = 0; sub < KC / 16; ++sub) {
            const int kloc = sub * 16;
            v8f c = {};   // lane = token col, vgpr r = code row r / r+8

            // A-frag (ISA 32-bit A 16x4): lane&15 = M (code row),
            // lane-half selects K pair (0,1 vs 2,3) across the 2 VGPRs.
            const float* arow = buf + (kloc + lo) * ROWP;
#pragma unroll
            for (int j = 0; j < DIM / 4; ++j) {
                v2f a = *(const v2f*)(arow + 4 * j + koff);
                c = __builtin_amdgcn_wmma_f32_16x16x4_f32(
                        /*neg_a=*/false, a, /*neg_b=*/false, bfrag[j],
                        /*c_mod=*/(short)0, c,
                        /*reuse_a=*/false, /*reuse_b=*/false);
            }

            // score = ||c||^2 - 2*dot ; running argmin (8 codes per lane)
#pragma unroll
            for (int r = 0; r < 8; ++r) {
                int   code = kc + kloc + r + 8 * hi;
                float s    = lds_cn[code] - 2.0f * c[r];
                if (s < best) { best = s; bidx = code; }
            }
        }

        wait_async0();      // next buffer's fill (this wave) complete
        __syncthreads();    // all waves done computing + all fills visible
    }

    // ---- Combine the two half-wave partials for each token ---------------
    float ob = __shfl_xor(best, 16, 32);
    int   oi = __shfl_xor(bidx, 16, 32);
    if (ob < best || (ob == best && oi < bidx)) { best = ob; bidx = oi; }

    // ---- Gather: lane pair (t, t+16) copies codebook[bidx] (64 floats ea.)
    {
        const v4f* src = (const v4f*)(cb + (size_t)bidx * DIM + hi * 64);
        v4f*       dst = (v4f*)(out + (size_t)tok * DIM + hi * 64);
#pragma unroll
        for (int i = 0; i < 16; ++i) dst[i] = src[i];
    }
}

extern "C" void kernel_launch(void* const* d_in, const int* in_sizes, int n_in,
                              void* d_out, int out_size, void* d_ws, size_t ws_size,
                              hipStream_t stream) {
    (void)in_sizes; (void)n_in; (void)out_size; (void)d_ws; (void)ws_size;
    const float* x  = (const float*)d_in[0];   // [32768, 128]
    const float* cb = (const float*)d_in[1];   // [1024, 128]
    float*       o  = (float*)d_out;           // [32768, 128]

    vq_argmin_gather<<<N_TOK / TOK_PER_WG, TPB, 0, stream>>>(x, cb, o);
}